// BaselineSVMModel_70119636074681
// MI455X (gfx1250) — compile-verified
//
#include <hip/hip_runtime.h>
#include <stdint.h>

typedef float v2f __attribute__((ext_vector_type(2)));
typedef float v8f __attribute__((ext_vector_type(8)));
typedef int   v4i_b __attribute__((vector_size(16)));                 // GCC vector, matches builtin
typedef __attribute__((address_space(1))) v4i_b* as1_v4i;             // global ptr to int4
typedef __attribute__((address_space(3))) v4i_b* as3_v4i;             // LDS ptr to int4

#if __has_builtin(__builtin_amdgcn_global_load_async_to_lds_b128) && \
    __has_builtin(__builtin_amdgcn_s_wait_asynccnt)
#define SP_USE_ASYNC_LDS 1
#else
#define SP_USE_ASYNC_LDS 0
#endif

static constexpr int kNumSp = 1024;
static constexpr int kC     = 21;          // classes
static constexpr int kF     = 256;         // feature dim
static constexpr int kPix   = 512 * 512;   // pixels
static constexpr int kRowF  = 36;          // 32 cols + 4 pad: conflict-free, 144B row stride
static constexpr int kChunk = 32;          // K staged per round

// workspace layout (floats)
static constexpr int kWfragOff = 0;                       // 64 ksteps x 32 lanes x float4
static constexpr int kProjOff  = 64 * 32 * 4;             // 8192: 1024 x 21 projected sums
static constexpr int kCntOff   = kProjOff + kNumSp * kC;  // 1024 counts

// Pre-swizzle W into per-(kstep,lane) float4 B-fragments; zero accumulators.
__global__ void sp_prep(const float* __restrict__ w, float* __restrict__ ws) {
  const int i = blockIdx.x * 256 + threadIdx.x;
  if (i < kNumSp * kC + kNumSp) ws[kProjOff + i] = 0.0f;
  if (i < 64 * 32) {
    const int s = i >> 5, l = i & 31, lo = l & 15, hi = l >> 4;
    const int k0 = 4 * s + 2 * hi;
    float4 f;
    f.x = w[lo * kF + k0];
    f.y = w[lo * kF + k0 + 1];
    f.z = (16 + lo < kC) ? w[(16 + lo) * kF + k0] : 0.0f;
    f.w = (16 + lo < kC) ? w[(16 + lo) * kF + k0 + 1] : 0.0f;
    *(float4*)(ws + kWfragOff + (size_t)i * 4) = f;
  }
}

// 32 pixels per wave: async-stage features to LDS, project with f32 WMMA
// (two pixel tiles x two class tiles share one b128 W fragment), scatter
// 21 atomics/pixel into the L2-resident accumulator.
__global__ __launch_bounds__(256) void sp_proj(const float* __restrict__ feats,
                                               const int* __restrict__ sp,
                                               float* __restrict__ ws) {
  __shared__ float ftile[8 * 32 * kRowF];  // 8 waves x (32 x 36) floats = 36.9 KB

  const int tid  = threadIdx.x;
  const int wid  = tid >> 5;   // wave id (wave32)
  const int lane = tid & 31;
  const int lo   = lane & 15;
  const int hi   = lane >> 4;
  const int rSub = lane >> 3;        // staging: 4 rows / instruction
  const int cSub = (lane & 7) * 4;   // staging: 8 lanes x float4 = 128B row segment

  float* tile = &ftile[wid * 32 * kRowF];
  const float4* __restrict__ Wf = (const float4*)(ws + kWfragOff);
  float* proj = ws + kProjOff;
  float* cnt  = ws + kCntOff;

  const int nT     = kPix / 32;      // 8192 tiles
  const int stride = gridDim.x * 8;  // waves in grid
  for (int t = blockIdx.x * 8 + wid; t < nT; t += stride) {
    const int p0 = t * 32;
    const float* fbase = feats + (size_t)p0 * kF;
    v8f c1a = {0.f,0.f,0.f,0.f,0.f,0.f,0.f,0.f};  // pix 0-15,  cls 0-15
    v8f c2a = {0.f,0.f,0.f,0.f,0.f,0.f,0.f,0.f};  // pix 0-15,  cls 16-20
    v8f c1b = {0.f,0.f,0.f,0.f,0.f,0.f,0.f,0.f};  // pix 16-31, cls 0-15
    v8f c2b = {0.f,0.f,0.f,0.f,0.f,0.f,0.f,0.f};  // pix 16-31, cls 16-20

    for (int kc = 0; kc < kF; kc += kChunk) {
#if SP_USE_ASYNC_LDS
#pragma unroll
      for (int j = 0; j < 8; ++j) {
        const int row = 4 * j + rSub;
        const float* gp = fbase + (size_t)row * kF + kc + cSub;
        float* lp = tile + row * kRowF + cSub;
        __builtin_amdgcn_global_load_async_to_lds_b128(
            (as1_v4i)(uintptr_t)gp,
            (as3_v4i)(uint32_t)(uintptr_t)lp,   // flat LDS addr low 32 bits == LDS offset
            0, 0);
      }
      __builtin_amdgcn_s_wait_asynccnt(0);
#else
#pragma unroll
      for (int j = 0; j < 8; ++j) {
        const int row = 4 * j + rSub;
        const float4 v = *(const float4*)(fbase + (size_t)row * kF + kc + cSub);
        *(float4*)(tile + row * kRowF + cSub) = v;
      }
#endif
#pragma unroll
      for (int ks = 0; ks < kChunk; ks += 4) {
        const v2f a1 = *(const v2f*)(tile + lo * kRowF + ks + 2 * hi);
        const v2f a2 = *(const v2f*)(tile + (16 + lo) * kRowF + ks + 2 * hi);
        const float4 bf = Wf[(size_t)(((kc + ks) >> 2) * 32 + lane)];
        const v2f b1 = {bf.x, bf.y};
        const v2f b2 = {bf.z, bf.w};
        c1a = __builtin_amdgcn_wmma_f32_16x16x4_f32(false, a1, false, b1, (short)0, c1a, false, false);
        c2a = __builtin_amdgcn_wmma_f32_16x16x4_f32(false, a1, false, b2, (short)0, c2a, false, false);
        c1b = __builtin_amdgcn_wmma_f32_16x16x4_f32(false, a2, false, b1, (short)0, c1b, false, false);
        c2b = __builtin_amdgcn_wmma_f32_16x16x4_f32(false, a2, false, b2, (short)0, c2b, false, false);
      }
    }

    // D layout: lane l, vgpr r -> pixel base + r + 8*(l>>4), class (l&15) [+16].
    const int pbA = p0 + 8 * hi;
    const int pbB = p0 + 16 + 8 * hi;
    const int4 sA0 = *(const int4*)(sp + pbA);
    const int4 sA1 = *(const int4*)(sp + pbA + 4);
    const int4 sB0 = *(const int4*)(sp + pbB);
    const int4 sB1 = *(const int4*)(sp + pbB + 4);
    const int sa[8] = {sA0.x, sA0.y, sA0.z, sA0.w, sA1.x, sA1.y, sA1.z, sA1.w};
    const int sb[8] = {sB0.x, sB0.y, sB0.z, sB0.w, sB1.x, sB1.y, sB1.z, sB1.w};
#pragma unroll
    for (int r = 0; r < 8; ++r) {
      atomicAdd(proj + sa[r] * kC + lo, c1a[r]);
      atomicAdd(proj + sb[r] * kC + lo, c1b[r]);
      if (lo < kC - 16) {
        atomicAdd(proj + sa[r] * kC + 16 + lo, c2a[r]);
        atomicAdd(proj + sb[r] * kC + 16 + lo, c2b[r]);
      }
    }
    if (lane < 16) {
      atomicAdd(cnt + sp[p0 + lane], 1.0f);
      atomicAdd(cnt + sp[p0 + 16 + lane], 1.0f);
    }
  }
}

__global__ void sp_final(const float* __restrict__ ws, float* __restrict__ out) {
  const int i = blockIdx.x * 256 + threadIdx.x;
  if (i < kNumSp * kC) {
    const float c = ws[kCntOff + i / kC];
    out[i] = ws[kProjOff + i] / fmaxf(c, 1.0f);
  }
}

extern "C" void kernel_launch(void* const* d_in, const int* in_sizes, int n_in,
                              void* d_out, int out_size, void* d_ws, size_t ws_size,
                              hipStream_t stream) {
  (void)in_sizes; (void)n_in; (void)out_size; (void)ws_size;
  const float* feats = (const float*)d_in[0];
  const int*   sp    = (const int*)d_in[1];
  const float* w     = (const float*)d_in[2];
  float* out = (float*)d_out;
  float* ws  = (float*)d_ws;

  const int prepN = kNumSp * kC + kNumSp;  // 22528 (also covers 2048 W-fragment threads)
  sp_prep<<<(prepN + 255) / 256, 256, 0, stream>>>(w, ws);
  sp_proj<<<1024, 256, 0, stream>>>(feats, sp, ws);
  sp_final<<<(kNumSp * kC + 255) / 256, 256, 0, stream>>>(ws, out);
}